// ANMP_Layer_44470091383041
// MI455X (gfx1250) — compile-verified
//
#include <hip/hip_runtime.h>
#include <math.h>

#define DIM 128
#define H   8
#define DH  16

typedef __attribute__((ext_vector_type(2))) float v2f;
typedef __attribute__((ext_vector_type(8))) float v8f;

// ---- ordered-uint encoding of float so unsigned atomicMax == float max ----
__device__ __forceinline__ unsigned f32_ord(float f) {
    unsigned u = __float_as_uint(f);
    return (u >> 31) ? ~u : (u | 0x80000000u);
}
__device__ __forceinline__ float ord_f32(unsigned u) {
    return (u >> 31) ? __uint_as_float(u & 0x7fffffffu) : __uint_as_float(~u);
}
// f32_ord(-FLT_MAX) == 0x00800000u

// =====================================================================
// C[n][d] = sum_k A[n][k] * W[d][k] + bias[d]     (A: N x 128, W: 128 x 128)
// Block = 256 threads = 8 waves; each wave owns a 16x128 C tile.
// W staged in LDS transposed, packed (k-pair, d) float2 -> conflict-free
// ds_load_b64 B fragments. A rows are CLAMPED to N-1 instead of guarded:
// a WMMA A-row only feeds the same C row, and out-of-range rows are never
// stored, so the inner loop has zero divergence.
// =====================================================================
__global__ __launch_bounds__(256)
void gemm_proj_kernel(const float* __restrict__ A, const float* __restrict__ W,
                      const float* __restrict__ bias, float* __restrict__ C, int N) {
    __shared__ float2 Bsh[64 * 128];                       // 64 KB
    const int tid = threadIdx.x;
    const float2* W2 = reinterpret_cast<const float2*>(W); // W row = 64 float2
#pragma unroll
    for (int i = 0; i < 32; ++i) {
        int idx = tid + i * 256;
        int k2 = idx >> 7;        // k-pair 0..63
        int d  = idx & 127;
        Bsh[idx] = W2[d * 64 + k2];   // (W[d][2k2], W[d][2k2+1])
    }
    __syncthreads();

    const int wave = tid >> 5;
    const int lane = tid & 31;
    const int half = lane >> 4;
    const int m    = lane & 15;
    const int rowBase = blockIdx.x * 128 + wave * 16;

    v8f acc[8] = {};

    const int arow = min(rowBase + m, N - 1);   // clamp: no divergence
    const float* Arow = A + (size_t)arow * DIM;
    __builtin_prefetch(Arow + 64, 0, 3);        // second half of the row

    for (int kt = 0; kt < 32; ++kt) {           // K = 128 = 32 * 4
        float2 av = *reinterpret_cast<const float2*>(Arow + kt * 4 + 2 * half);
        v2f afrag; afrag.x = av.x; afrag.y = av.y;
        const float2* brow = &Bsh[(2 * kt + half) * 128 + m];
#pragma unroll
        for (int nt = 0; nt < 8; ++nt) {
            float2 bv = brow[nt * 16];
            v2f bfrag; bfrag.x = bv.x; bfrag.y = bv.y;
            acc[nt] = __builtin_amdgcn_wmma_f32_16x16x4_f32(
                false, afrag, false, bfrag, (short)0, acc[nt], false, false);
        }
    }
#pragma unroll
    for (int nt = 0; nt < 8; ++nt) {
        int d = nt * 16 + m;
        float bb = bias[d];
#pragma unroll
        for (int r = 0; r < 8; ++r) {
            int row = rowBase + r + 8 * half;   // C layout: M = r + 8*half
            if (row < N) C[(size_t)row * DIM + d] = acc[nt][r] + bb;
        }
    }
}

// =====================================================================
// Final: x = query + (msg/denom) @ Wp^T + bp ; LayerNorm(x) * g + b
// =====================================================================
__global__ __launch_bounds__(256)
void gemm_out_ln_kernel(const float* __restrict__ msg, const float* __restrict__ denom,
                        const float* __restrict__ W, const float* __restrict__ bp,
                        const float* __restrict__ query, const float* __restrict__ ln_g,
                        const float* __restrict__ ln_b, float* __restrict__ out, int N) {
    __shared__ float2 Bsh[64 * 128];
    const int tid = threadIdx.x;
    const float2* W2 = reinterpret_cast<const float2*>(W);
#pragma unroll
    for (int i = 0; i < 32; ++i) {
        int idx = tid + i * 256;
        Bsh[idx] = W2[(idx & 127) * 64 + (idx >> 7)];
    }
    __syncthreads();

    const int wave = tid >> 5;
    const int lane = tid & 31;
    const int half = lane >> 4;
    const int m    = lane & 15;
    const int rowBase = blockIdx.x * 128 + wave * 16;

    v8f acc[8] = {};

    const int arow = min(rowBase + m, N - 1);   // clamp: no divergence
    const float* Mrow = msg   + (size_t)arow * DIM;
    const float* Drow = denom + (size_t)arow * H;
    __builtin_prefetch(Mrow + 64, 0, 3);

    for (int kt = 0; kt < 32; ++kt) {
        int k = kt * 4 + 2 * half;
        float2 mv = *reinterpret_cast<const float2*>(Mrow + k);
        float dn = Drow[k >> 4];                // head = k/16 (pair stays in head)
        dn = (dn == 0.0f) ? 1.0f : dn;
        float inv = 1.0f / dn;
        v2f afrag; afrag.x = mv.x * inv; afrag.y = mv.y * inv;
        const float2* brow = &Bsh[(2 * kt + half) * 128 + m];
#pragma unroll
        for (int nt = 0; nt < 8; ++nt) {
            float2 bv = brow[nt * 16];
            v2f bfrag; bfrag.x = bv.x; bfrag.y = bv.y;
            acc[nt] = __builtin_amdgcn_wmma_f32_16x16x4_f32(
                false, afrag, false, bfrag, (short)0, acc[nt], false, false);
        }
    }

    float gd[8], bd[8], bpv[8];
#pragma unroll
    for (int nt = 0; nt < 8; ++nt) {
        int d = nt * 16 + m;
        gd[nt] = ln_g[d]; bd[nt] = ln_b[d]; bpv[nt] = bp[d];
    }
#pragma unroll
    for (int r = 0; r < 8; ++r) {
        int row  = rowBase + r + 8 * half;
        int lrow = min(row, N - 1);             // clamp loads; guard stores only
        float x[8];
        float s1 = 0.0f, s2 = 0.0f;
#pragma unroll
        for (int nt = 0; nt < 8; ++nt) {
            float q = query[(size_t)lrow * DIM + nt * 16 + m];
            float v = acc[nt][r] + bpv[nt] + q;
            x[nt] = v; s1 += v; s2 += v * v;
        }
        // reduce the 16 lanes of this half (row spans 8 accsets x 16 lanes)
#pragma unroll
        for (int off = 8; off >= 1; off >>= 1) {
            s1 += __shfl_xor(s1, off);
            s2 += __shfl_xor(s2, off);
        }
        float mu   = s1 * (1.0f / DIM);
        float var  = s2 * (1.0f / DIM) - mu * mu;
        float rstd = rsqrtf(var + 1e-5f);
        if (row < N) {
#pragma unroll
            for (int nt = 0; nt < 8; ++nt)
                out[(size_t)row * DIM + nt * 16 + m] = (x[nt] - mu) * rstd * gd[nt] + bd[nt];
        }
    }
}

// =====================================================================
// Per-edge score: e[edge][h] = sum_dh a[h][dh]*prelu(Qp[qi]+Kp[ki]) ; seg-max
// One wave per edge; lane owns 4 dims (4 lanes per head).
// =====================================================================
__global__ __launch_bounds__(256)
void edge_score_kernel(const float* __restrict__ Qp, const float* __restrict__ Kp,
                       const int* __restrict__ qidx, const int* __restrict__ kidx,
                       const float* __restrict__ a, const float* __restrict__ prelu_w,
                       float* __restrict__ e_out, unsigned* __restrict__ maxq,
                       long long E) {
    long long wid = ((long long)blockIdx.x * blockDim.x + threadIdx.x) >> 5;
    if (wid >= E) return;
    const int lane = threadIdx.x & 31;
    const int qi = qidx[wid], ki = kidx[wid];
    const float pw = prelu_w[0];
    const float4 qv = *reinterpret_cast<const float4*>(Qp + (size_t)qi * DIM + lane * 4);
    const float4 kv = *reinterpret_cast<const float4*>(Kp + (size_t)ki * DIM + lane * 4);
    const float4 av = *reinterpret_cast<const float4*>(a + lane * 4);
    float s = 0.0f, t;
    t = qv.x + kv.x; s += av.x * (t >= 0.0f ? t : pw * t);
    t = qv.y + kv.y; s += av.y * (t >= 0.0f ? t : pw * t);
    t = qv.z + kv.z; s += av.z * (t >= 0.0f ? t : pw * t);
    t = qv.w + kv.w; s += av.w * (t >= 0.0f ? t : pw * t);
    s += __shfl_xor(s, 1);
    s += __shfl_xor(s, 2);                     // sum over the head's 4 lanes
    const int head = lane >> 2;
    if ((lane & 3) == 0) {
        e_out[(size_t)wid * H + head] = s;
        atomicMax(&maxq[(size_t)qi * H + head], f32_ord(s));
    }
}

// =====================================================================
// Softmax weight + scatter: denom[q][h] += w ; msg[q][:] += w * Vp[ki][:]
// =====================================================================
__global__ __launch_bounds__(256)
void edge_aggregate_kernel(const float* __restrict__ Vp, const int* __restrict__ qidx,
                           const int* __restrict__ kidx, const float* __restrict__ e_in,
                           const unsigned* __restrict__ maxq, float* __restrict__ denom,
                           float* __restrict__ msg, long long E) {
    long long wid = ((long long)blockIdx.x * blockDim.x + threadIdx.x) >> 5;
    if (wid >= E) return;
    const int lane = threadIdx.x & 31;
    const int qi = qidx[wid], ki = kidx[wid];
    const int head = lane >> 2;
    const float ev = e_in[(size_t)wid * H + head];
    const float mq = ord_f32(maxq[(size_t)qi * H + head]);
    const float w  = expf(ev - mq);
    if ((lane & 3) == 0) atomicAdd(&denom[(size_t)qi * H + head], w);
    const float4 vv = *reinterpret_cast<const float4*>(Vp + (size_t)ki * DIM + lane * 4);
    float* mrow = msg + (size_t)qi * DIM + lane * 4;
    atomicAdd(mrow + 0, w * vv.x);
    atomicAdd(mrow + 1, w * vv.y);
    atomicAdd(mrow + 2, w * vv.z);
    atomicAdd(mrow + 3, w * vv.w);
}

__global__ __launch_bounds__(256)
void init_kernel(unsigned* __restrict__ maxq, float* __restrict__ denom,
                 float* __restrict__ msg, int NQ) {
    int i = blockIdx.x * blockDim.x + threadIdx.x;
    if (i < NQ * DIM) msg[i] = 0.0f;
    if (i < NQ * H) { denom[i] = 0.0f; maxq[i] = 0x00800000u; }  // ord(-FLT_MAX)
}

extern "C" void kernel_launch(void* const* d_in, const int* in_sizes, int n_in,
                              void* d_out, int out_size, void* d_ws, size_t ws_size,
                              hipStream_t stream) {
    (void)n_in; (void)out_size; (void)ws_size;
    const float* query  = (const float*)d_in[0];
    const float* keys   = (const float*)d_in[1];
    const float* values = (const float*)d_in[2];
    const int*   qidx   = (const int*)d_in[3];
    const int*   kidx   = (const int*)d_in[4];
    const float* Wq = (const float*)d_in[5];  const float* bq = (const float*)d_in[6];
    const float* Wk = (const float*)d_in[7];  const float* bk = (const float*)d_in[8];
    const float* Wv = (const float*)d_in[9];  const float* bv = (const float*)d_in[10];
    const float* Wp = (const float*)d_in[11]; const float* bp = (const float*)d_in[12];
    const float* a       = (const float*)d_in[13];
    const float* prelu_w = (const float*)d_in[14];
    const float* ln_g    = (const float*)d_in[15];
    const float* ln_b    = (const float*)d_in[16];

    const int NQ = in_sizes[0] / DIM;
    const int NK = in_sizes[1] / DIM;
    const long long E = in_sizes[3];

    // workspace carve-out (~265 MB)
    float* ws = (float*)d_ws;
    float* Qp = ws;  ws += (size_t)NQ * DIM;
    float* Kp = ws;  ws += (size_t)NK * DIM;
    float* Vp = ws;  ws += (size_t)NK * DIM;
    float* ev = ws;  ws += (size_t)E * H;
    unsigned* maxq = (unsigned*)ws; ws += (size_t)NQ * H;
    float* denom = ws; ws += (size_t)NQ * H;
    float* msg   = ws; ws += (size_t)NQ * DIM;

    {
        int n = NQ * DIM;
        init_kernel<<<(n + 255) / 256, 256, 0, stream>>>(maxq, denom, msg, NQ);
    }
    const int gq = (NQ + 127) / 128;
    const int gk = (NK + 127) / 128;
    gemm_proj_kernel<<<gq, 256, 0, stream>>>(query,  Wq, bq, Qp, NQ);
    gemm_proj_kernel<<<gk, 256, 0, stream>>>(keys,   Wk, bk, Kp, NK);
    gemm_proj_kernel<<<gk, 256, 0, stream>>>(values, Wv, bv, Vp, NK);

    const long long ethreads = E * 32;           // one wave per edge
    const int eblocks = (int)((ethreads + 255) / 256);
    edge_score_kernel<<<eblocks, 256, 0, stream>>>(Qp, Kp, qidx, kidx, a, prelu_w,
                                                   ev, maxq, E);
    edge_aggregate_kernel<<<eblocks, 256, 0, stream>>>(Vp, qidx, kidx, ev, maxq,
                                                       denom, msg, E);
    gemm_out_ln_kernel<<<gq, 256, 0, stream>>>(msg, denom, Wp, bp, query,
                                               ln_g, ln_b, (float*)d_out, NQ);
}